// SageGnNetwork_50371376447773
// MI455X (gfx1250) — compile-verified
//
#include <hip/hip_runtime.h>
#include <hip/hip_bf16.h>

#define N_NODES_C 50000
#define N_EDGES_C 800000
#define DIM 128
#define NCLS 24
#define RT 5   // row tiles per wave: 625 * (16*5) == 50000 exactly

typedef float v8f __attribute__((ext_vector_type(8)));
typedef float v2f __attribute__((ext_vector_type(2)));

// ---------------- degree / deg_inv ----------------

__global__ void deg_kernel(const long long* __restrict__ dst, float* __restrict__ deg, int nE) {
    int e = blockIdx.x * blockDim.x + threadIdx.x;
    if (e < nE) unsafeAtomicAdd(&deg[dst[e]], 1.0f);
}

__global__ void deginv_kernel(float* __restrict__ deg, int n) {
    int i = blockIdx.x * blockDim.x + threadIdx.x;
    if (i < n) deg[i] = 1.0f / fmaxf(deg[i], 1.0f);
}

// ---------------- edge scatter (segment_sum) ----------------
// blockDim = (32, 8): lane = float4 chunk (32 x 16B = 512B row), threadIdx.y = edge in block.
__global__ void scatter_kernel(const float* __restrict__ x, const long long* __restrict__ src,
                               const long long* __restrict__ dst, float* __restrict__ msg, int nE) {
    int chunk = threadIdx.x;
    int e = blockIdx.x * blockDim.y + threadIdx.y;
    if (e >= nE) return;
    long long s = src[e], d = dst[e];
    const float4* xr = (const float4*)(x + (size_t)s * DIM);
    float4 v = xr[chunk];
    float* mp = msg + (size_t)d * DIM + (size_t)chunk * 4;
    unsafeAtomicAdd(mp + 0, v.x);
    unsafeAtomicAdd(mp + 1, v.y);
    unsafeAtomicAdd(mp + 2, v.z);
    unsafeAtomicAdd(mp + 3, v.w);
}

// ---------------- fused SAGE layer GEMM via V_WMMA_F32_16X16X4_F32 ----------------
// out[n][j] = relu( (msg[n]*dinv[n]) . Wl[j] + bl[j] + xin[n] . Wr[j] )
// One wave computes RT row-tiles (16*RT rows) x one 16-col tile, reusing the
// Wl/Wr B-fragments across all RT row tiles (loads/WMMA = 1.2 instead of 2.0).
// A-frag (16x4 f32): lanes 0-15 -> K={0,1}, lanes 16-31 -> K={2,3}.
// C/D: VGPR v, lanes 0-15: M=v, lanes 16-31: M=v+8; N = lane&15.
// Requires nNodes % (16*RT) == 0 (50000 = 625 * 80): no row guards.
template <int OUTD>
__global__ void __launch_bounds__(256)
sage_gemm_kernel(const float* __restrict__ msg, const float* __restrict__ dinv,
                 const float* __restrict__ xin, const float* __restrict__ Wl,
                 const float* __restrict__ bl, const float* __restrict__ Wr,
                 float* __restrict__ out) {
    const int lane = threadIdx.x;         // 0..31
    const int m    = lane & 15;
    const int hi   = lane >> 4;           // 0 or 1
    const int koff = hi * 2;
    const int row0 = blockIdx.x * (16 * RT);
    const int col0 = threadIdx.y * 16;

    // B-side weight row (= output column j) for this lane; mask only for OUTD=24
    const int jB    = col0 + m;
    const int jc    = (OUTD == DIM) ? jB : (jB < OUTD ? jB : OUTD - 1);
    const float wsc = (OUTD == DIM) ? 1.0f : (jB < OUTD ? 1.0f : 0.0f);
    const float* wlrow = Wl + (size_t)jc * DIM + koff;
    const float* wrrow = Wr + (size_t)jc * DIM + koff;

    const float* mrow[RT];
    const float* xrow[RT];
    float dv[RT];
#pragma unroll
    for (int t = 0; t < RT; ++t) {
        int r = row0 + t * 16 + m;
        mrow[t] = msg + (size_t)r * DIM + koff;
        xrow[t] = xin + (size_t)r * DIM + koff;
        dv[t]   = dinv[r];
    }

    v8f acc[RT] = {};

#pragma unroll 2
    for (int k0 = 0; k0 < DIM; k0 += 4) {
        v2f b1 = *(const v2f*)(wlrow + k0);
        v2f b2 = *(const v2f*)(wrrow + k0);
        if (OUTD != DIM) { b1 *= wsc; b2 *= wsc; }
#pragma unroll
        for (int t = 0; t < RT; ++t) {
            v2f a1 = *(const v2f*)(mrow[t] + k0);
            a1 *= dv[t];                           // fold deg_inv into aggregated term
            acc[t] = __builtin_amdgcn_wmma_f32_16x16x4_f32(false, a1, false, b1,
                                                           (short)0, acc[t], false, false);
            v2f a2 = *(const v2f*)(xrow[t] + k0);
            acc[t] = __builtin_amdgcn_wmma_f32_16x16x4_f32(false, a2, false, b2,
                                                           (short)0, acc[t], false, false);
        }
    }

    const int cc   = col0 + m;
    const float bv = (cc < OUTD) ? bl[cc] : 0.0f;
#pragma unroll
    for (int t = 0; t < RT; ++t) {
#pragma unroll
        for (int v = 0; v < 8; ++v) {
            int r = row0 + t * 16 + v + hi * 8;
            if (cc < OUTD) {                       // compile-time true for OUTD==DIM
                float val = acc[t][v] + bv;
                out[(size_t)r * OUTD + cc] = fmaxf(val, 0.0f);
            }
        }
    }
}

// ---------------- row-wise log_softmax over 24 classes (one wave32 per row) ----------------
__global__ void logsoftmax_kernel(const float* __restrict__ h, float* __restrict__ out, int n) {
    int lane = threadIdx.x;
    int row = blockIdx.x * blockDim.y + threadIdx.y;
    if (row >= n) return;
    const float NEG_INF = -__builtin_inff();
    float v = (lane < NCLS) ? h[(size_t)row * NCLS + lane] : NEG_INF;
    float mx = v;
#pragma unroll
    for (int o = 16; o > 0; o >>= 1) mx = fmaxf(mx, __shfl_xor(mx, o, 32));
    float ex = (lane < NCLS) ? __expf(v - mx) : 0.0f;
    float sm = ex;
#pragma unroll
    for (int o = 16; o > 0; o >>= 1) sm += __shfl_xor(sm, o, 32);
    float ls = __logf(sm);
    if (lane < NCLS) out[(size_t)row * NCLS + lane] = v - mx - ls;
}

// ---------------- host ----------------

extern "C" void kernel_launch(void* const* d_in, const int* in_sizes, int n_in,
                              void* d_out, int out_size, void* d_ws, size_t ws_size,
                              hipStream_t stream) {
    const float* x        = (const float*)d_in[0];
    const long long* ei   = (const long long*)d_in[1];
    const long long* src  = ei;
    const long long* dst  = ei + N_EDGES_C;
    const float* Wl1 = (const float*)d_in[2];
    const float* bl1 = (const float*)d_in[3];
    const float* Wr1 = (const float*)d_in[4];
    const float* Wlm = (const float*)d_in[5];
    const float* blm = (const float*)d_in[6];
    const float* Wrm = (const float*)d_in[7];
    const float* Wl5 = (const float*)d_in[8];
    const float* bl5 = (const float*)d_in[9];
    const float* Wr5 = (const float*)d_in[10];

    float* ws  = (float*)d_ws;
    float* deg = ws;                                // 50000 (deg, then deg_inv in place)
    float* msg = deg + 50048;                       // N x 128
    float* hA  = msg + (size_t)N_NODES_C * DIM;     // N x 128
    float* hB  = hA  + (size_t)N_NODES_C * DIM;     // N x 128
    float* h5  = hB  + (size_t)N_NODES_C * DIM;     // N x 24

    hipMemsetAsync(deg, 0, (size_t)N_NODES_C * sizeof(float), stream);
    deg_kernel<<<(N_EDGES_C + 255) / 256, 256, 0, stream>>>(dst, deg, N_EDGES_C);
    deginv_kernel<<<(N_NODES_C + 255) / 256, 256, 0, stream>>>(deg, N_NODES_C);

    const dim3 sblk(32, 8);
    const int  sgrid = (N_EDGES_C + 7) / 8;
    const int  ggrid = N_NODES_C / (16 * RT);       // 625

    auto layer128 = [&](const float* hin, const float* Wl, const float* bl,
                        const float* Wr, float* hout) {
        hipMemsetAsync(msg, 0, (size_t)N_NODES_C * DIM * sizeof(float), stream);
        scatter_kernel<<<sgrid, sblk, 0, stream>>>(hin, src, dst, msg, N_EDGES_C);
        sage_gemm_kernel<DIM><<<dim3(ggrid), dim3(32, DIM / 16), 0, stream>>>(
            msg, deg, hin, Wl, bl, Wr, hout);
    };

    layer128(x,  Wl1,                 bl1,           Wr1,                 hA);
    layer128(hA, Wlm + 0 * DIM * DIM, blm + 0 * DIM, Wrm + 0 * DIM * DIM, hB);
    layer128(hB, Wlm + 1 * DIM * DIM, blm + 1 * DIM, Wrm + 1 * DIM * DIM, hA);
    layer128(hA, Wlm + 2 * DIM * DIM, blm + 2 * DIM, Wrm + 2 * DIM * DIM, hB);

    // conv5: 128 -> 24 (2 column tiles of 16, stores masked to 24)
    hipMemsetAsync(msg, 0, (size_t)N_NODES_C * DIM * sizeof(float), stream);
    scatter_kernel<<<sgrid, sblk, 0, stream>>>(hB, src, dst, msg, N_EDGES_C);
    sage_gemm_kernel<NCLS><<<dim3(ggrid), dim3(32, 2), 0, stream>>>(
        msg, deg, hB, Wl5, bl5, Wr5, h5);

    logsoftmax_kernel<<<N_NODES_C / 8, dim3(32, 8), 0, stream>>>(h5, (float*)d_out, N_NODES_C);
}